// CenterLoss_82927228551479
// MI455X (gfx1250) — compile-verified
//
#include <hip/hip_runtime.h>

typedef __attribute__((ext_vector_type(2))) float v2f;
typedef __attribute__((ext_vector_type(8))) float v8f;

#define D_DIM 128

// ---------------------------------------------------------------------------
// Kernel 1: c_sq[c] = sum_d centers[c][d]^2   (one wave per class row)
// ---------------------------------------------------------------------------
__global__ void csq_kernel(const float* __restrict__ centers,
                           float* __restrict__ csq, int C) {
    int wave = threadIdx.x >> 5;
    int lane = threadIdx.x & 31;
    int row  = blockIdx.x * (blockDim.x >> 5) + wave;
    if (row >= C) return;
    const float* p = centers + (size_t)row * D_DIM;
    float s = 0.f;
    #pragma unroll
    for (int k = 0; k < D_DIM; k += 32) { float v = p[k + lane]; s += v * v; }
    #pragma unroll
    for (int off = 16; off; off >>= 1) s += __shfl_xor(s, off, 32);
    if (lane == 0) csq[row] = s;
}

// ---------------------------------------------------------------------------
// Kernel 2: pre[b] = 1 + ||e_b - centers[t_b]||^2 - ||e_b||^2  (wave per row)
// ---------------------------------------------------------------------------
__global__ void pre_kernel(const float* __restrict__ e,
                           const int* __restrict__ targets,
                           const float* __restrict__ centers,
                           float* __restrict__ pre, int B) {
    int wave = threadIdx.x >> 5;
    int lane = threadIdx.x & 31;
    int row  = blockIdx.x * (blockDim.x >> 5) + wave;
    if (row >= B) return;
    int t = targets[row];
    const float* pe = e + (size_t)row * D_DIM;
    const float* pc = centers + (size_t)t * D_DIM;
    float se = 0.f, si = 0.f;
    #pragma unroll
    for (int k = 0; k < D_DIM; k += 32) {
        float ev = pe[k + lane];
        float cv = pc[k + lane];
        se += ev * ev;
        float d = ev - cv;
        si += d * d;
    }
    #pragma unroll
    for (int off = 16; off; off >>= 1) {
        se += __shfl_xor(se, off, 32);
        si += __shfl_xor(si, off, 32);
    }
    if (lane == 0) pre[row] = 1.0f + si - se;
}

// ---------------------------------------------------------------------------
// Kernel 3: tiled fp32 WMMA GEMM cross = E * centers^T, fused masked-relu sum.
// Block = 256 threads (8 waves) covering 128 rows; wave w owns a 16-row tile.
// Loop over 63 column tiles with DOUBLE-BUFFERED LDS staging (one barrier per
// tile, next tile's global loads issued before the WMMA chain) and all 32
// B-fragments preloaded to registers so the WMMA chain never drains DScnt.
// ---------------------------------------------------------------------------
#define LDS_STRIDE 132  // 128 + 4 pad: conflict-free, keeps 16B alignment

__launch_bounds__(256)
__global__ void tile_kernel(const float* __restrict__ e,
                            const int* __restrict__ targets,
                            const float* __restrict__ centers,
                            const float* __restrict__ csq,
                            const float* __restrict__ pre,
                            float* __restrict__ partials,
                            int C) {
    __shared__ float ldsB[2][16 * LDS_STRIDE];
    __shared__ float wsum[8];

    const int tid  = threadIdx.x;
    const int wave = tid >> 5;
    const int lane = tid & 31;
    const int h    = lane >> 4;   // half-wave: selects K pair / M offset
    const int idx  = lane & 15;   // M (for A) / N (for B,C)

    const int waveRowBase = blockIdx.x * 128 + wave * 16;

    // --- A fragments: ISA 16x4 f32 layout. Lane (h,idx): row=idx,
    //     chunk kk covers K = 4*kk + 2*h + {0,1}. 32 chunks = full D=128.
    const float* arow = e + (size_t)(waveRowBase + idx) * D_DIM + 2 * h;
    v2f afrag[32];
    #pragma unroll
    for (int kk = 0; kk < 32; ++kk)
        afrag[kk] = *(const v2f*)(arow + 4 * kk);

    // --- Per-lane epilogue metadata: C/D layout puts M = i + 8*h in vgpr i.
    float prev[8];
    int   tgt[8];
    #pragma unroll
    for (int i = 0; i < 8; ++i) {
        int r   = waveRowBase + i + 8 * h;
        prev[i] = pre[r];
        tgt[i]  = targets[r];
    }

    // Staging role of this thread: class cc (0..15), dims [d0, d0+8).
    const int cc = tid >> 4;
    const int d0 = (tid & 15) * 8;
    const int nColTiles = (C + 15) >> 4;  // 63

    // Prologue: stage tile 0 into buffer 0.
    float4 sx0, sx1;
    {
        int gcls = cc;
        if (gcls > C - 1) gcls = C - 1;
        const float4* g = (const float4*)(centers + (size_t)gcls * D_DIM + d0);
        sx0 = g[0];
        sx1 = g[1];
        float* dst = &ldsB[0][cc * LDS_STRIDE + d0];
        *(float4*)(dst)     = sx0;
        *(float4*)(dst + 4) = sx1;
    }
    __syncthreads();

    float lsum = 0.f;
    int p = 0;
    for (int ct = 0; ct < nColTiles; ++ct, p ^= 1) {
        const bool hasNext = (ct + 1) < nColTiles;

        // Issue next tile's global loads early (latency hidden under WMMAs).
        if (hasNext) {
            int gcls = ((ct + 1) << 4) + cc;
            if (gcls > C - 1) gcls = C - 1;
            const float4* g = (const float4*)(centers + (size_t)gcls * D_DIM + d0);
            sx0 = g[0];
            sx1 = g[1];
        }

        // Preload all 32 B fragments for this tile (16 x ds_load_2addr_b64).
        const float* bcol = &ldsB[p][idx * LDS_STRIDE + 2 * h];
        v2f bfrag[32];
        #pragma unroll
        for (int kk = 0; kk < 32; ++kk)
            bfrag[kk] = *(const v2f*)(bcol + 4 * kk);

        // --- 32 x V_WMMA_F32_16X16X4_F32, two accumulator chains for ILP.
        v8f acc0 = {}, acc1 = {};
        #pragma unroll
        for (int kk = 0; kk < 32; kk += 2) {
            acc0 = __builtin_amdgcn_wmma_f32_16x16x4_f32(
                false, afrag[kk],     false, bfrag[kk],     (short)0, acc0,
                false, false);
            acc1 = __builtin_amdgcn_wmma_f32_16x16x4_f32(
                false, afrag[kk + 1], false, bfrag[kk + 1], (short)0, acc1,
                false, false);
        }
        v8f acc = acc0 + acc1;

        // Store next tile into the other LDS buffer.
        if (hasNext) {
            float* dst = &ldsB[p ^ 1][cc * LDS_STRIDE + d0];
            *(float4*)(dst)     = sx0;
            *(float4*)(dst + 4) = sx1;
        }

        // --- Fused epilogue: term = relu(pre[m] - c_sq[n] + 2*cross[m][n])
        int  col      = (ct << 4) + idx;
        bool colValid = col < C;
        float cs = csq[colValid ? col : (C - 1)];
        #pragma unroll
        for (int i = 0; i < 8; ++i) {
            float t  = prev[i] - cs + 2.0f * acc[i];
            bool use = colValid && (t > 0.f) && (tgt[i] != col);
            lsum += use ? t : 0.f;
        }

        // Single barrier per tile: readers of buf p are done, and writers of
        // buf p^1 have completed their ds_stores.
        __syncthreads();
    }

    // --- Deterministic block reduction.
    #pragma unroll
    for (int off = 16; off; off >>= 1) lsum += __shfl_xor(lsum, off, 32);
    if (lane == 0) wsum[wave] = lsum;
    __syncthreads();
    if (wave == 0) {
        float s = (lane < 8) ? wsum[lane] : 0.f;
        #pragma unroll
        for (int off = 4; off; off >>= 1) s += __shfl_xor(s, off, 32);
        if (lane == 0) partials[blockIdx.x] = s;
    }
}

// ---------------------------------------------------------------------------
// Kernel 4: fixed-order final reduction in double, scale, write scalar.
// ---------------------------------------------------------------------------
__global__ void finalize_kernel(const float* __restrict__ partials, int n,
                                float* __restrict__ out, double scale) {
    __shared__ double sh[256];
    double s = 0.0;
    for (int i = threadIdx.x; i < n; i += 256) s += (double)partials[i];
    sh[threadIdx.x] = s;
    __syncthreads();
    for (int off = 128; off; off >>= 1) {
        if ((int)threadIdx.x < off) sh[threadIdx.x] += sh[threadIdx.x + off];
        __syncthreads();
    }
    if (threadIdx.x == 0) out[0] = (float)(sh[0] * scale);
}

// ---------------------------------------------------------------------------
extern "C" void kernel_launch(void* const* d_in, const int* in_sizes, int n_in,
                              void* d_out, int out_size, void* d_ws, size_t ws_size,
                              hipStream_t stream) {
    const float* e       = (const float*)d_in[0];
    const int*   targets = (const int*)d_in[1];
    const float* centers = (const float*)d_in[2];
    float*       out     = (float*)d_out;

    const int B = in_sizes[0] / D_DIM;   // 32768
    const int C = in_sizes[2] / D_DIM;   // 1000

    float* ws       = (float*)d_ws;
    float* csq      = ws;                // 1024 slots (C <= 1024)
    float* pre      = ws + 1024;         // B floats
    float* partials = pre + B;           // B/128 floats

    // 1) per-class squared norms
    csq_kernel<<<(C + 3) / 4, 128, 0, stream>>>(centers, csq, C);
    // 2) per-row pre = 1 + intra - e_sq
    pre_kernel<<<(B + 7) / 8, 256, 0, stream>>>(e, targets, centers, pre, B);
    // 3) WMMA GEMM + fused masked relu partial sums
    const int nBlocks = B / 128;         // 256
    tile_kernel<<<nBlocks, 256, 0, stream>>>(e, targets, centers, csq, pre,
                                             partials, C);
    // 4) deterministic final reduce + scale
    const double scale = 1.0 / ((double)B * (double)(C - 1));
    finalize_kernel<<<1, 256, 0, stream>>>(partials, nBlocks, out, scale);
}